// SpecialSpmm_4277787427326
// MI455X (gfx1250) — compile-verified
//
#include <hip/hip_runtime.h>
#include <cstdint>
#include <cstddef>

// ---------------------------------------------------------------------------
// COO SpMM for MI455X (gfx1250):
//   out[row[e], :] += values[e] * b[col[e], :]   with D = 64 features.
//
// Memory-bound (0.24 FLOP/byte): WMMA is deliberately NOT used (no dense
// accumulation structure; it would add 16x redundant FLOPs + layout shuffles
// and save zero bytes). Instead we use the CDNA5 async-tensor path:
//   - global_load_async_to_lds_b128 : double-buffered row gathers into LDS
//     (ASYNCcnt, no VGPR return, deep outstanding queue per wave)
//   - s_wait_asynccnt               : split-wait releasing exactly the older
//     batch (async loads retire in order)
//   - global_atomic_add_f32 scope:SCOPE_DEV : native no-return f32 atomics
//     performed at L2 (device scope: different WGPs hit the same rows);
//     lane l owns features l and l+32 so each atomic instruction covers a
//     fully coalesced 128B span.
//   - nontemporal loads (th:TH_LOAD_NT) for the once-streamed edge metadata
//     so it does not evict the L2-resident b/out working set (51 MB in the
//     192 MB L2, reused ~16x per row).
// ---------------------------------------------------------------------------

#define D_FEAT            64
#define BATCH             8                     // edges (rows) per wave-batch
#define WAVES_PER_BLOCK   8
#define BLOCK_THREADS     (WAVES_PER_BLOCK * 32)
#define ROW_BYTES         (D_FEAT * 4)          // 256 B per gathered row
#define BUF_BYTES         (BATCH * ROW_BYTES)   // 2048 B per batch buffer
#define WAVE_LDS_BYTES    (2 * BUF_BYTES)       // double buffered: 4096 B
#define WAVE_LDS_FLOATS   (WAVE_LDS_BYTES / 4)  // 1024 floats per wave

// Issue 4 async B128 loads that gather BATCH=8 rows of b into this wave's
// LDS buffer. Each instruction: 32 lanes x 16B = 512B = 2 rows.
// Lane mapping per j: lanes 0-15 -> edge (2j), lanes 16-31 -> edge (2j+1);
// (lane&15) selects the 16B chunk within the 256B row.
__device__ __forceinline__ void async_gather_batch(
    const float* __restrict__ bmat, const int* __restrict__ coli,
    int eb, int E, unsigned ldsBufAddr, int lane)
{
  int ei = eb + (lane & 7);
  if (ei >= E) ei = E - 1;          // tail clamp (redundant gather, harmless)
  const int c8  = __builtin_nontemporal_load(&coli[ei]);  // lanes 0..7 valid
  const int sub = lane & 15;
#pragma unroll
  for (int j = 0; j < 4; ++j) {
    int ce = __shfl(c8, 2 * j + (lane >> 4), 32);
    const float* gp = bmat + (size_t)ce * D_FEAT + sub * 4;   // 16B chunk
    unsigned lp = ldsBufAddr + (unsigned)(j * 512 + lane * 16);
    asm volatile("global_load_async_to_lds_b128 %0, %1, off"
                 :: "v"(lp), "v"(gp)
                 : "memory");
  }
}

__global__ __launch_bounds__(BLOCK_THREADS)
void SpecialSpmm_coo_async_kernel(
    const int*   __restrict__ rowi,
    const int*   __restrict__ coli,
    const float* __restrict__ vals,
    const float* __restrict__ bmat,
    float*       __restrict__ out,
    int E)
{
  __shared__ __align__(16) float smem[WAVES_PER_BLOCK * WAVE_LDS_FLOATS];

  const int lane = threadIdx.x & 31;
  const int wv   = threadIdx.x >> 5;

  // Low 32 bits of a flat shared pointer == LDS byte address (aperture rule).
  const unsigned wldsAddr = (unsigned)(uintptr_t)(&smem[wv * WAVE_LDS_FLOATS]);

  const int nBatches = (E + BATCH - 1) / BATCH;
  const int stride   = gridDim.x * WAVES_PER_BLOCK;
  int bk = blockIdx.x * WAVES_PER_BLOCK + wv;
  if (bk >= nBatches) return;

  int buf = 0;
  // Prologue: start gathers for our first batch.
  async_gather_batch(bmat, coli, bk * BATCH, E, wldsAddr, lane);

  while (true) {
    const int eb = bk * BATCH;

    // Per-batch metadata: lanes 0..7 hold row/val for the 8 edges.
    int ei = eb + (lane & 7);
    const bool ok = (ei < E);
    if (!ok) ei = E - 1;
    const int   r8 = __builtin_nontemporal_load(&rowi[ei]);
    const float v8 = ok ? __builtin_nontemporal_load(&vals[ei]) : 0.0f;

    const int nxt = bk + stride;
    if (nxt < nBatches) {
      // Prefetch next batch into the other buffer, then wait until only
      // those 4 newest async ops remain outstanding (in-order retirement
      // => our current batch's 4 gathers have landed in LDS).
      async_gather_batch(bmat, coli, nxt * BATCH, E,
                         wldsAddr + (unsigned)((buf ^ 1) * BUF_BYTES), lane);
      asm volatile("s_wait_asynccnt 0x4" ::: "memory");
    } else {
      asm volatile("s_wait_asynccnt 0x0" ::: "memory");
    }

    // Consume current buffer: lane l handles features l and l+32 of each of
    // the 8 gathered rows, scaling and scatter-adding with device-scope
    // no-return f32 atomics (RMW performed at L2).
    const int base = wv * WAVE_LDS_FLOATS + buf * (BUF_BYTES / 4);
#pragma unroll
    for (int i = 0; i < BATCH; ++i) {
      const int   r = __shfl(r8, i, 32);
      const float v = __shfl(v8, i, 32);
      const float f0 = smem[base + i * D_FEAT + lane];        // ds_load_b32
      const float f1 = smem[base + i * D_FEAT + 32 + lane];
      const float a0 = v * f0;
      const float a1 = v * f1;
      float* p0 = out + (size_t)r * D_FEAT + lane;            // coalesced 128B
      float* p1 = p0 + 32;                                    // coalesced 128B
      asm volatile("global_atomic_add_f32 %0, %1, off scope:SCOPE_DEV"
                   :: "v"(p0), "v"(a0) : "memory");
      asm volatile("global_atomic_add_f32 %0, %1, off scope:SCOPE_DEV"
                   :: "v"(p1), "v"(a1) : "memory");
    }

    if (nxt >= nBatches) break;
    bk  = nxt;
    buf ^= 1;
  }
}

extern "C" void kernel_launch(void* const* d_in, const int* in_sizes, int n_in,
                              void* d_out, int out_size, void* d_ws, size_t ws_size,
                              hipStream_t stream)
{
  // setup_inputs() order: indices [2,E] int32, values [E] f32, N scalar, b [N,64] f32
  const int*   indices = (const int*)d_in[0];
  const float* values  = (const float*)d_in[1];
  const float* bmat    = (const float*)d_in[3];
  const int    E       = in_sizes[1];

  const int* rowi = indices;       // indices[0, :]
  const int* coli = indices + E;   // indices[1, :]
  float* out = (float*)d_out;

  // Atomic-accumulation kernel: output must start at zero every launch.
  hipMemsetAsync(d_out, 0, (size_t)out_size * sizeof(float), stream);

  const int nBatches = (E + BATCH - 1) / BATCH;
  int blocks = (nBatches + WAVES_PER_BLOCK - 1) / WAVES_PER_BLOCK;
  if (blocks > 4096) blocks = 4096;   // grid-stride: ~6 batches/wave -> the
                                      // double-buffered async pipeline is live
  if (blocks < 1) blocks = 1;

  SpecialSpmm_coo_async_kernel<<<blocks, BLOCK_THREADS, 0, stream>>>(
      rowi, coli, values, bmat, out, E);
}